// LabelLoss_73512660239077
// MI455X (gfx1250) — compile-verified
//
#include <hip/hip_runtime.h>
#include <cstdint>

// Problem constants from the reference: B,J,C,H,W = 16,16,64,256,256
#define B_    16
#define J_    16
#define C_    64
#define HW_   65536          // H*W
#define SPLIT_ 4             // split each channel's argmax across 4 blocks
#define SEG_  (HW_ / SPLIT_) // 16384 floats per block

// ---------------------------------------------------------------------------
// Kernel 1: partial argmax over one SEG_ slice of one (b,j) heatmap channel.
// Streams data through LDS with double-buffered GLOBAL_LOAD_ASYNC_TO_LDS_B128
// (CDNA5 async-memory path, tracked by ASYNCcnt).
//
// Layout per block (256 threads = 8 wave32):
//   wave w owns floats [w*2048, (w+1)*2048) of the segment,
//   processed as 4 chunks of 512 floats; each chunk = 4 async B128 issues
//   (32 lanes x 16B = 128 floats per issue).
//   LDS staging: 8 waves x 2 buffers x 512 floats = 32 KB.
// ---------------------------------------------------------------------------
__global__ __launch_bounds__(256) void argmax_part_kernel(
    const float* __restrict__ hm, float* __restrict__ wsval,
    int* __restrict__ wsidx) {
  __shared__ float stage[8192];  // 32 KB async staging
  __shared__ float rv[256];
  __shared__ int   ri[256];

  const int tid  = threadIdx.x;
  const int lane = tid & 31;   // wave32
  const int wave = tid >> 5;
  const int blk  = blockIdx.x;
  const int s    = blk & (SPLIT_ - 1);
  const int bj   = blk >> 2;

  const float* base = hm + (size_t)bj * HW_ + (size_t)s * SEG_;
  const int wbase = wave * 2048;          // float offset of this wave's slice

  // Prologue: issue chunk 0 into buffer 0 (ASYNCcnt -> 4)
#pragma unroll
  for (int q = 0; q < 4; ++q) {
    unsigned lds_b =
        (unsigned)(uintptr_t)(const void*)&stage[wave * 1024 + q * 128 + lane * 4];
    unsigned long long g =
        (unsigned long long)(uintptr_t)(base + wbase + q * 128 + lane * 4);
    asm volatile("global_load_async_to_lds_b128 %0, %1, off"
                 :: "v"(lds_b), "v"(g) : "memory");
  }

  float maxv = -__builtin_inff();
  int   maxi = 0;

  for (int t = 0; t < 4; ++t) {
    if (t < 3) {
      // Issue next chunk into the other buffer (ASYNCcnt -> 8), then wait
      // until only those 4 newest remain: async loads complete in order,
      // so cnt<=4 means the buffer we are about to read is resident.
      const int tn = t + 1;
      const int bufn = tn & 1;
#pragma unroll
      for (int q = 0; q < 4; ++q) {
        unsigned lds_b = (unsigned)(uintptr_t)(const void*)
            &stage[wave * 1024 + bufn * 512 + q * 128 + lane * 4];
        unsigned long long g = (unsigned long long)(uintptr_t)
            (base + wbase + tn * 512 + q * 128 + lane * 4);
        asm volatile("global_load_async_to_lds_b128 %0, %1, off"
                     :: "v"(lds_b), "v"(g) : "memory");
      }
      asm volatile("s_wait_asynccnt 0x4" ::: "memory");
    } else {
      asm volatile("s_wait_asynccnt 0x0" ::: "memory");
    }

    const int buf = t & 1;
#pragma unroll
    for (int q = 0; q < 4; ++q) {
      const float4 v = *reinterpret_cast<const float4*>(
          &stage[wave * 1024 + buf * 512 + q * 128 + lane * 4]);
      const int gi = s * SEG_ + wbase + t * 512 + q * 128 + lane * 4;
      const float vv[4] = {v.x, v.y, v.z, v.w};
#pragma unroll
      for (int e = 0; e < 4; ++e) {
        // strict '>' keeps the first (lowest-index) maximum, matching argmax
        if (vv[e] > maxv) { maxv = vv[e]; maxi = gi + e; }
      }
    }
  }

  // Block tree-reduction on (max value, min index on ties)
  rv[tid] = maxv;
  ri[tid] = maxi;
  __syncthreads();
  for (int off = 128; off > 0; off >>= 1) {
    if (tid < off) {
      const float v2 = rv[tid + off];
      const int   i2 = ri[tid + off];
      if (v2 > rv[tid] || (v2 == rv[tid] && i2 < ri[tid])) {
        rv[tid] = v2;
        ri[tid] = i2;
      }
    }
    __syncthreads();
  }
  if (tid == 0) {
    wsval[blk] = rv[0];
    wsidx[blk] = ri[0];  // already a full [0, HW) index (includes s*SEG_)
  }
}

// ---------------------------------------------------------------------------
// Kernel 2: combine split partials -> final argmax idx per (b,j); gather
// pred[b, :, idx], squared-diff vs gt, mean over (j,c) -> out[b].
// ---------------------------------------------------------------------------
__global__ __launch_bounds__(256) void gather_mse_kernel(
    const float* __restrict__ pred, const float* __restrict__ gt,
    const float* __restrict__ wsval, const int* __restrict__ wsidx,
    float* __restrict__ out) {
  __shared__ int   jidx[J_];
  __shared__ float red[256];

  const int b   = blockIdx.x;
  const int tid = threadIdx.x;

  if (tid < J_) {
    const int basee = (b * J_ + tid) * SPLIT_;
    float bv = wsval[basee];
    int   bi = wsidx[basee];
    // ascending-s scan with strict '>' keeps the lowest index on value ties
    for (int s2 = 1; s2 < SPLIT_; ++s2) {
      const float v = wsval[basee + s2];
      if (v > bv) { bv = v; bi = wsidx[basee + s2]; }
    }
    jidx[tid] = bi;
  }
  __syncthreads();

  float acc = 0.0f;
  for (int e = tid; e < J_ * C_; e += 256) {
    const int j = e >> 6;         // C_ == 64
    const int c = e & (C_ - 1);
    const float g = pred[((size_t)b * C_ + c) * HW_ + jidx[j]];
    const float d = g - gt[(b * J_ + j) * C_ + c];
    acc += d * d;
  }

  red[tid] = acc;
  __syncthreads();
  for (int off = 128; off > 0; off >>= 1) {
    if (tid < off) red[tid] += red[tid + off];
    __syncthreads();
  }
  if (tid == 0) out[b] = red[0] * (1.0f / (float)(J_ * C_));
}

// ---------------------------------------------------------------------------
extern "C" void kernel_launch(void* const* d_in, const int* in_sizes, int n_in,
                              void* d_out, int out_size, void* d_ws,
                              size_t ws_size, hipStream_t stream) {
  (void)in_sizes; (void)n_in; (void)out_size; (void)ws_size;
  const float* pred = (const float*)d_in[0];  // [B,C,H,W]
  const float* gt   = (const float*)d_in[1];  // [B,J,C]
  const float* hm   = (const float*)d_in[2];  // [B,J,H,W]

  float* wsval = (float*)d_ws;                                   // 1024 floats
  int*   wsidx = (int*)((char*)d_ws + B_ * J_ * SPLIT_ * sizeof(float));

  argmax_part_kernel<<<B_ * J_ * SPLIT_, 256, 0, stream>>>(hm, wsval, wsidx);
  gather_mse_kernel<<<B_, 256, 0, stream>>>(pred, gt, wsval, wsidx,
                                            (float*)d_out);
}